// ProcessFeatures_12051678233388
// MI455X (gfx1250) — compile-verified
//
#include <hip/hip_runtime.h>
#include <hip/hip_bf16.h>

typedef float v2f __attribute__((ext_vector_type(2)));
typedef float v8f __attribute__((ext_vector_type(8)));

#define ELEMS_PER_IMG 262144   // 64*64*64
#define LDA 68                 // padded LDS row stride (bank-conflict free, 16B aligned)
#define LDP 65                 // padded P-tile stride

// ---------------------------------------------------------------------------
// Kernel 1: per-pair 64x64xK=4096 GEMM on the f32 WMMA pipe, fused wrapped-
// diagonal reduction + first-occurrence argmax.
// grid = 1024 (i*32+j), block = 256 (8 wave32s)
//
// A[h][m]   = sat[i,c,h,dx]        (m = c*64+dx)
// B[m][dy]  = grd[j,c,dy,63-dx]    -> staged into LDS pre-reversed so the
//                                     fragment pair {B[k][n],B[k+1][n]} is an
//                                     ascending contiguous ds_load_b64.
// ---------------------------------------------------------------------------
__global__ void __launch_bounds__(256)
corr_argmax_kernel(const float* __restrict__ sat, const float* __restrict__ grd,
                   int* __restrict__ orien_ws, float* __restrict__ orien_out)
{
    __shared__ float As[64 * LDA];   // As[h][dx]   = sat[i,c,h,dx]
    __shared__ float Gs[64 * LDA];   // Gs[dy][k]   = grd[j,c,dy,63-k]  (reversed)
    __shared__ float Pt[64 * LDP];   // P[h][dy]
    __shared__ float corr1d[64];

    const int pair = blockIdx.x;
    const int i = pair >> 5, j = pair & 31;
    const float* Sb = sat + (size_t)i * ELEMS_PER_IMG;
    const float* Gb = grd + (size_t)j * ELEMS_PER_IMG;

    const int t    = threadIdx.x;
    const int lane = t & 31;
    const int wave = t >> 5;
    const int tm   = wave & 3;          // M tile (0..3)
    const int tn0  = (wave >> 2) * 2;   // N tiles tn0, tn0+1
    const int hf   = lane >> 4;         // lane half (0/1)
    const int l16  = lane & 15;

    // cooperative copy: thread t -> row t/4, 16 floats at column (t%4)*16
    const int cr = t >> 2;
    const int cq = (t & 3) * 16;

    v8f acc0 = {};
    v8f acc1 = {};

    for (int c = 0; c < 64; ++c) {
        const float* sg = Sb + c * 4096;   // contiguous 64x64 slice
        const float* gg = Gb + c * 4096;
        __syncthreads();
        {
            const float4* s4 = (const float4*)(sg + cr * 64 + cq);
            const float4* g4 = (const float4*)(gg + cr * 64 + cq);
            float4* ad = (float4*)&As[cr * LDA + cq];        // 16B aligned
            ad[0] = s4[0]; ad[1] = s4[1]; ad[2] = s4[2]; ad[3] = s4[3];
            // reversed store: Gs[dy][k] = grd row element (63-k)
            // global float4 at dxp=cq+4u -> LDS float4 at col 60-cq-4u, lanes flipped
#pragma unroll
            for (int u = 0; u < 4; ++u) {
                float4 g = g4[u];
                float4 r; r.x = g.w; r.y = g.z; r.z = g.y; r.w = g.x;
                *(float4*)&Gs[cr * LDA + (60 - cq - 4 * u)] = r;   // 16B aligned
            }
        }
        __syncthreads();

        const int arow = tm * 16 + l16;
        const int n0   = tn0 * 16 + l16;
        const int n1   = n0 + 16;
#pragma unroll
        for (int ks = 0; ks < 16; ++ks) {
            const int kk = ks * 4 + hf * 2;     // this half's K offset
            // A 16x4 frag: lane = M row; {A[row][kk], A[row][kk+1]}
            v2f a  = *(const v2f*)&As[arow * LDA + kk];
            // B 4x16 frag: lane%16 = N col; {B[kk][n], B[kk+1][n]} = Gs[n][kk..kk+1]
            v2f b0 = *(const v2f*)&Gs[n0 * LDA + kk];
            v2f b1 = *(const v2f*)&Gs[n1 * LDA + kk];
            acc0 = __builtin_amdgcn_wmma_f32_16x16x4_f32(false, a, false, b0,
                                                         (short)0, acc0, false, false);
            acc1 = __builtin_amdgcn_wmma_f32_16x16x4_f32(false, a, false, b1,
                                                         (short)0, acc1, false, false);
        }
    }

    __syncthreads();
    // C/D layout: VGPR v, lane -> row = v + 8*half, col = lane%16
#pragma unroll
    for (int v = 0; v < 8; ++v) {
        const int row = tm * 16 + v + hf * 8;
        Pt[row * LDP + tn0 * 16 + l16]      = acc0[v];
        Pt[row * LDP + tn0 * 16 + 16 + l16] = acc1[v];
    }
    __syncthreads();

    // corr_out[y] = sum_dy P[(y+dy)%64][dy]
    if (t < 64) {
        float s = 0.f;
        for (int dy = 0; dy < 64; ++dy)
            s += Pt[((t + dy) & 63) * LDP + dy];
        corr1d[t] = s;
    }
    __syncthreads();
    if (t == 0) {
        float best = corr1d[0];
        int bi = 0;
        for (int y = 1; y < 64; ++y)
            if (corr1d[y] > best) { best = corr1d[y]; bi = y; }   // first max
        orien_ws[pair]  = bi;
        orien_out[pair] = (float)bi;
    }
}

// ---------------------------------------------------------------------------
// Kernel 2: norm[i] = sqrt(sum(sat[i]^2) + 1e-8)   (independent of j)
// ---------------------------------------------------------------------------
__global__ void __launch_bounds__(256)
sumsq_kernel(const float* __restrict__ sat, float* __restrict__ norms)
{
    __shared__ float red[256];
    const int i = blockIdx.x;
    const float4* p4 = (const float4*)(sat + (size_t)i * ELEMS_PER_IMG);
    float s = 0.f;
    for (int idx = threadIdx.x; idx < ELEMS_PER_IMG / 4; idx += 256) {
        float4 v = p4[idx];
        s += v.x * v.x + v.y * v.y + v.z * v.z + v.w * v.w;
    }
    red[threadIdx.x] = s;
    __syncthreads();
    for (int off = 128; off > 0; off >>= 1) {
        if (threadIdx.x < off) red[threadIdx.x] += red[threadIdx.x + off];
        __syncthreads();
    }
    if (threadIdx.x == 0) norms[i] = sqrtf(red[0] + 1e-8f);
}

// ---------------------------------------------------------------------------
// Kernel 3: dot at the winning shift, distance[j,i] = 2 - 2*dot/norm[i]
// grid = 1024 pairs, block = 256
// ---------------------------------------------------------------------------
__global__ void __launch_bounds__(256)
dot_kernel(const float* __restrict__ sat, const float* __restrict__ grd,
           const int* __restrict__ orien_ws, const float* __restrict__ norms,
           float* __restrict__ dist_out)
{
    __shared__ float red[256];
    const int pair = blockIdx.x;
    const int i = pair >> 5, j = pair & 31;
    const int o = orien_ws[pair];
    const float* Sb = sat + (size_t)i * ELEMS_PER_IMG;
    const float* Gb = grd + (size_t)j * ELEMS_PER_IMG;

    float s = 0.f;
    // 4096 rows of 64; dot(sat row circularly shifted by o, grd row)
    for (int r = threadIdx.x; r < 4096; r += 256) {
        const float* sr = Sb + r * 64;
        const float* gr = Gb + r * 64;
        int k = 0;
        for (; k < 64 - o; ++k) s += sr[k + o] * gr[k];        // contiguous part 1
        for (; k < 64; ++k)     s += sr[k + o - 64] * gr[k];   // wrapped part 2
    }
    red[threadIdx.x] = s;
    __syncthreads();
    for (int off = 128; off > 0; off >>= 1) {
        if (threadIdx.x < off) red[threadIdx.x] += red[threadIdx.x + off];
        __syncthreads();
    }
    if (threadIdx.x == 0) {
        float dot = red[0] / norms[i];
        dist_out[j * 32 + i] = 2.f - 2.f * dot;   // distance is [Bg,Bs] = dot.T
    }
}

// ---------------------------------------------------------------------------
// Launch: out = [ sat (8388608) | grd (8388608) | distance (1024) | orien (1024) ]
// ---------------------------------------------------------------------------
extern "C" void kernel_launch(void* const* d_in, const int* in_sizes, int n_in,
                              void* d_out, int out_size, void* d_ws, size_t ws_size,
                              hipStream_t stream)
{
    const float* sat = (const float*)d_in[0];
    const float* grd = (const float*)d_in[1];
    float* out = (float*)d_out;

    const size_t N = (size_t)32 * ELEMS_PER_IMG;   // 8388608
    float* out_dist  = out + 2 * N;
    float* out_orien = out + 2 * N + 1024;

    int*   ws_orien = (int*)d_ws;
    float* ws_norm  = (float*)((char*)d_ws + 1024 * sizeof(int));

    // passthrough outputs
    hipMemcpyAsync(out,     sat, N * sizeof(float), hipMemcpyDeviceToDevice, stream);
    hipMemcpyAsync(out + N, grd, N * sizeof(float), hipMemcpyDeviceToDevice, stream);

    corr_argmax_kernel<<<1024, 256, 0, stream>>>(sat, grd, ws_orien, out_orien);
    sumsq_kernel<<<32, 256, 0, stream>>>(sat, ws_norm);
    dot_kernel<<<1024, 256, 0, stream>>>(sat, grd, ws_orien, ws_norm, out_dist);
}